// LocalFeatureAggregation_41832981463399
// MI455X (gfx1250) — compile-verified
//
#include <hip/hip_runtime.h>

typedef __attribute__((ext_vector_type(2))) float v2f;
typedef __attribute__((ext_vector_type(8))) float v8f;
typedef __attribute__((ext_vector_type(4))) int   v4i;

#define KNN_K 16
#define DIN   64

// ds_swizzle SWAPX16: offset[15]=0 (group-of-32), xor_mask=0x10, or=0, and=0x1f
__device__ __forceinline__ float swapx16(float x) {
    return __builtin_bit_cast(float,
        __builtin_amdgcn_ds_swizzle(__builtin_bit_cast(int, x), 0x401f));
}

__global__ __launch_bounds__(256) void lfa_wmma_kernel(
    const float* __restrict__ inputs,   // [N, 64]
    const float* __restrict__ geo,      // [N, 16, 4]
    const int*   __restrict__ knn,      // [N, 16]
    const float* __restrict__ W,        // [64, 4]
    const float* __restrict__ bias,     // [64]
    float* __restrict__ out,            // [N, 128]
    int n_points)
{
    const int lane = threadIdx.x & 31;
    const int wave = threadIdx.x >> 5;
    const int n    = blockIdx.x * 8 + wave;   // wave-uniform -> EXEC all-1s for WMMA
    if (n >= n_points) return;

    const int half = lane >> 4;   // 0: lanes 0-15 (K=0,1) ; 1: lanes 16-31 (K=2,3)
    const int l16  = lane & 15;

    // All offsets kept in 32-bit elements (max 12.8M) -> saddr+voffset global loads
    v2f a = *(const v2f*)(geo + n * (KNN_K * 4) + l16 * 4 + half * 2);

    const float* wp = W    + l16 * 4 + half * 2;  // per-tile stride: 64 elem = imm offset
    const float* bp = bias + l16;

    float xs[4];
    #pragma unroll
    for (int t = 0; t < 4; ++t) {
        v2f bm = *(const v2f*)(wp + t * 64);
        const float bb = bp[t * 16];
        v8f c = { bb, bb, bb, bb, bb, bb, bb, bb };

        v8f d = __builtin_amdgcn_wmma_f32_16x16x4_f32(
            false, a, false, bm, (short)0, c, false, false);

        // sum_k leaky(x,0.2) = 0.6*sum(x) + 0.4*sum(|x|); |x| is a free src modifier
        float S  = ((d[0] + d[1]) + (d[2] + d[3])) + ((d[4] + d[5]) + (d[6] + d[7]));
        float SA = ((fabsf(d[0]) + fabsf(d[1])) + (fabsf(d[2]) + fabsf(d[3])))
                 + ((fabsf(d[4]) + fabsf(d[5])) + (fabsf(d[6]) + fabsf(d[7])));
        float loc = fmaf(0.6f / KNN_K, S, (0.4f / KNN_K) * SA);  // local-half mean part
        xs[t] = loc + swapx16(loc);   // combine rows M=0-7 with M=8-15 (one ds_swizzle)
    }

    // every lane stores 2 distinct columns: half 0 -> tiles 0,1 ; half 1 -> tiles 2,3
    const float o0 = half ? xs[2] : xs[0];    // column half*32 + l16
    const float o1 = half ? xs[3] : xs[1];    // column half*32 + 16 + l16
    float* xout = out + n * 128 + half * 32 + l16;
    xout[0]  = o0;
    xout[16] = o1;

    // gather-mean: lane covers d = 2*lane, 2*lane+1 ; indices via 4x b128 loads
    const int* kn = knn + n * KNN_K;
    v4i idx[4];
    #pragma unroll
    for (int q = 0; q < 4; ++q)
        idx[q] = *(const v4i*)(kn + 4 * q);

    const int lane2 = lane << 1;
    v2f g = {0.f, 0.f};
    #pragma unroll
    for (int q = 0; q < 4; ++q) {
        #pragma unroll
        for (int j = 0; j < 4; ++j) {
            const int off = (idx[q][j] << 6) + lane2;   // 32-bit element offset
            g += *(const v2f*)(inputs + off);
        }
    }
    g *= (1.0f / KNN_K);
    *(v2f*)(out + n * 128 + 64 + lane2) = g;
}

extern "C" void kernel_launch(void* const* d_in, const int* in_sizes, int n_in,
                              void* d_out, int out_size, void* d_ws, size_t ws_size,
                              hipStream_t stream) {
    const float* inputs = (const float*)d_in[0];
    const float* geo    = (const float*)d_in[1];
    const int*   knn    = (const int*)d_in[2];
    const float* W      = (const float*)d_in[3];
    const float* bias   = (const float*)d_in[4];
    float* out          = (float*)d_out;

    const int n_points = in_sizes[0] / DIN;    // inputs is [N, 64]
    const int blocks   = (n_points + 7) / 8;   // 8 waves (points) per block

    hipLaunchKernelGGL(lfa_wmma_kernel, dim3(blocks), dim3(256), 0, stream,
                       inputs, geo, knn, W, bias, out, n_points);
}